// GHMLoss_3307124818276
// MI455X (gfx1250) — compile-verified
//
#include <hip/hip_runtime.h>
#include <hip/hip_bf16.h>
#include <math.h>

// ---------------- configuration ----------------
#define GHM_N      16777216
#define GHM_BINS   10
#define NBLK       1024          // partial-reduction blocks (deterministic slots)
#define TPB        256           // 8 wave32s per block
#define WAVES      (TPB / 32)
#define PROWS      16            // partial matrix rows (10 used, padded to 16 for WMMA)
#define MOMENTUM   0.75f

typedef float v2f __attribute__((ext_vector_type(2)));
typedef float v8f __attribute__((ext_vector_type(8)));

// ---------------- init: zero the partial buffers ----------------
__global__ void ghm_zero(float* p, int n) {
    int i = blockIdx.x * blockDim.x + threadIdx.x;
    if (i < n) p[i] = 0.0f;
}

// ---------------- main streaming pass (HBM-bound) ----------------
// outputs: [N,2] f32 ; targets: [N] i32 ; partials: [2][16][NBLK] f32
__device__ __forceinline__ void ghm_accum(float x0, float x1, int tv,
                                          float* hist, int wave, int lane) {
    float xt = tv ? x1 : x0;
    float m  = fmaxf(x0, x1);
    float lse = m + logf(expf(x0 - m) + expf(x1 - m));
    float logpt = xt - lse;                 // <= 0 (up to rounding)
    float p = expf(logpt);
    float g = 2.0f * (1.0f - p);
    if (g < 1.0f) {                         // valid sample
        int b = (int)floorf(g * 10.0f);
        b = b < 0 ? 0 : (b > GHM_BINS - 1 ? GHM_BINS - 1 : b);
        int base = (wave * GHM_BINS + b) * 32 + lane;        // private slot: no conflicts
        hist[base] += 1.0f;                                  // count
        hist[WAVES * GHM_BINS * 32 + base] += (-logpt);      // loss sum
    }
}

__global__ __launch_bounds__(TPB) void ghm_main(const float4* __restrict__ out4,
                                                const int2*  __restrict__ tgt2,
                                                float* __restrict__ partials) {
    __shared__ float hist[2 * WAVES * GHM_BINS * 32];   // 20 KB, [kind][wave][bin][lane]
    const int lane = threadIdx.x & 31;
    const int wave = threadIdx.x >> 5;

    for (int b = 0; b < GHM_BINS; ++b) {
        hist[(wave * GHM_BINS + b) * 32 + lane] = 0.0f;
        hist[WAVES * GHM_BINS * 32 + (wave * GHM_BINS + b) * 32 + lane] = 0.0f;
    }
    __syncthreads();

    const int M = GHM_N / 2;                            // float4 / int2 elements
    const int stride = gridDim.x * blockDim.x;
    for (int i = blockIdx.x * blockDim.x + threadIdx.x; i < M; i += stride) {
        float4 o = out4[i];                             // 2 samples, b128 load
        int2   t = tgt2[i];                             // 2 targets, b64 load
        ghm_accum(o.x, o.y, t.x, hist, wave, lane);
        ghm_accum(o.z, o.w, t.y, hist, wave, lane);
    }
    __syncthreads();

    // deterministic block reduction: 20 threads each own one (kind,bin)
    if (threadIdx.x < 2 * GHM_BINS) {
        int kind = threadIdx.x / GHM_BINS;
        int b    = threadIdx.x % GHM_BINS;
        float s = 0.0f;
        const float* h = hist + kind * WAVES * GHM_BINS * 32;
        for (int w = 0; w < WAVES; ++w)
            for (int l = 0; l < 32; ++l)
                s += h[(w * GHM_BINS + b) * 32 + l];
        partials[(kind * PROWS + b) * NBLK + blockIdx.x] = s;
    }
}

// ---------------- finalize: WMMA ones-vector reduction + scalar ----------------
// partials: counts = P[0..15][0..NBLK), sums = P[16..31][0..NBLK)
__global__ __launch_bounds__(32) void ghm_finalize(const float* __restrict__ partials,
                                                   const float* __restrict__ acc_sum,
                                                   float* __restrict__ out) {
    const int lane = threadIdx.x;
    const int row  = lane & 15;
    const int koff = (lane >> 4) * 2;   // lanes 0-15 supply K=0,1 ; lanes 16-31 K=2,3

    v8f accC = {};
    v8f accS = {};
    v2f bones; bones[0] = 1.0f; bones[1] = 1.0f;   // B = ones -> D[m,n] = row-sum(A row m)

    const float* Pc = partials;
    const float* Ps = partials + PROWS * NBLK;

    for (int c = 0; c < NBLK; c += 4) {
        v2f a;
        const float* pc = Pc + row * NBLK + c + koff;
        a[0] = pc[0]; a[1] = pc[1];
        accC = __builtin_amdgcn_wmma_f32_16x16x4_f32(false, a, false, bones,
                                                     (short)0, accC, false, false);
        const float* ps = Ps + row * NBLK + c + koff;
        a[0] = ps[0]; a[1] = ps[1];
        accS = __builtin_amdgcn_wmma_f32_16x16x4_f32(false, a, false, bones,
                                                     (short)0, accS, false, false);
    }

    // D layout: VGPR r, lanes 0-15 -> M=r ; lanes 16-31 -> M=8+r ; N = lane&15.
    // Column N=0 lives in lane 0 (rows 0-7) and lane 16 (rows 8-15).
    __shared__ float cnt[PROWS], sm[PROWS];
    if (lane == 0)  for (int r = 0; r < 8; ++r) { cnt[r]     = accC[r]; sm[r]     = accS[r]; }
    if (lane == 16) for (int r = 0; r < 8; ++r) { cnt[8 + r] = accC[r]; sm[8 + r] = accS[r]; }
    __syncthreads();

    if (lane == 0) {
        float res = 0.0f;
        for (int b = 0; b < GHM_BINS; ++b) {
            float cb = cnt[b];
            if (cb > 0.0f) {
                float na = MOMENTUM * acc_sum[b] + (1.0f - MOMENTUM) * cb;
                res += sm[b] / na;     // == mean(loss * weights)
            }
        }
        out[0] = res;
    }
}

// ---------------- launcher ----------------
extern "C" void kernel_launch(void* const* d_in, const int* in_sizes, int n_in,
                              void* d_out, int out_size, void* d_ws, size_t ws_size,
                              hipStream_t stream) {
    const float4* out4 = (const float4*)d_in[0];   // outputs [N,2] f32
    const int2*   tgt2 = (const int2*)d_in[1];     // targets [N] i32
    const float*  accs = (const float*)d_in[2];    // acc_sum [10] f32
    float* partials = (float*)d_ws;                // 2 * 16 * NBLK floats = 128 KB
    float* outp = (float*)d_out;

    const int nPart = 2 * PROWS * NBLK;
    ghm_zero<<<(nPart + TPB - 1) / TPB, TPB, 0, stream>>>(partials, nPart);
    ghm_main<<<NBLK, TPB, 0, stream>>>(out4, tgt2, partials);
    ghm_finalize<<<1, 32, 0, stream>>>(partials, accs, outp);
}